// RelativeLabelLoss_14319420965548
// MI455X (gfx1250) — compile-verified
//
#include <hip/hip_runtime.h>
#include <hip/hip_bf16.h>
#include <cstdint>

#define GAMMA_F   0.2f
#define C_DIM     100000
#define T_DIM     20
#define NWAVE     8
#define BLOCK     256
#define LANES     32
// Each wave owns C_DIM/NWAVE = 12500 floats = 3125 float4 (50000 bytes, 16B aligned).
#define WAVE_F4    3125
#define TILE_F4    128                 // float4 per wave-tile = 32 lanes * 4 unroll
#define NT         (WAVE_F4 / TILE_F4) // 24 full tiles
#define TAIL_START (NT * TILE_F4)      // 3072
#define TILE_BYTES (TILE_F4 * 16)      // 2048
#define WAVE_BYTES (WAVE_F4 * 16)      // 50000
#define NBUF       3                   // 3-stage async ring per wave

__device__ __forceinline__ float fexp(float x) {
#if __has_builtin(__builtin_amdgcn_exp2f)
  return __builtin_amdgcn_exp2f(x * 1.4426950408889634f);
#else
  return __expf(x);
#endif
}
__device__ __forceinline__ float flog(float x) {
#if __has_builtin(__builtin_amdgcn_logf)
  return __builtin_amdgcn_logf(x) * 0.6931471805599453f;
#else
  return __logf(x);
#endif
}

// Branchless online logsumexp update: exactly one transcendental per element.
__device__ __forceinline__ void lse_update(float& m, float& s, float v) {
  float d = v - m;                 // m starts at -inf: d=+inf -> e=0 -> s=1, m=v
  float e = fexp(-fabsf(d));
  bool  gt = d > 0.0f;
  s = gt ? (s * e + 1.0f) : (s + e);
  m = gt ? v : m;
}

__device__ __forceinline__ void lse_merge(float& m, float& s, float mo, float so) {
  float M = fmaxf(m, mo);
  s = s * fexp(m - M) + so * fexp(mo - M);
  m = M;
}

// Issue one 2KB wave-tile: 4 async b128 ops, streamed (non-temporal in L2).
#define ISSUE_TILE(LA_BASE, GO_BASE)                                          \
  do {                                                                        \
    uint32_t _la = (LA_BASE) + (uint32_t)(lane * 16);                         \
    uint32_t _go = (GO_BASE) + (uint32_t)(lane * 16);                         \
    _Pragma("unroll")                                                         \
    for (int _u = 0; _u < 4; ++_u) {                                          \
      asm volatile("global_load_async_to_lds_b128 %0, %1, %2 th:TH_LOAD_NT"   \
                   :: "v"(_la + _u * 512), "v"(_go + _u * 512), "s"(saddr)    \
                   : "memory");                                               \
    }                                                                         \
  } while (0)

extern "C" __global__ void __launch_bounds__(BLOCK)
rll_row_kernel(const float* __restrict__ x, const int* __restrict__ y,
               float* __restrict__ ws, int B) {
  // [3 buffers][8 waves][512 floats] = 48 KB streaming ring (of 320 KB/WGP)
  __shared__ __align__(16) float smem[NBUF * NWAVE * TILE_F4 * 4];
  __shared__ float wm[NWAVE], wss[NWAVE];

  const int row  = blockIdx.x;
  const int tid  = threadIdx.x;
  const int wave = tid >> 5;
  const int lane = tid & 31;

  const float*   rowp  = x + (size_t)row * C_DIM;
  const uint64_t saddr = (uint64_t)(uintptr_t)rowp;

  // Low 32 bits of a generic LDS address == LDS byte offset (aperture form).
  const uint32_t lds0  = (uint32_t)(uintptr_t)(&smem[0]);
  const uint32_t buf_b0 = lds0 + (uint32_t)(wave * TILE_BYTES);
  const uint32_t buf_b1 = buf_b0 + (uint32_t)(NWAVE * TILE_BYTES);
  const uint32_t buf_b2 = buf_b1 + (uint32_t)(NWAVE * TILE_BYTES);
  const uint32_t gbase  = (uint32_t)(wave * WAVE_BYTES);

  const float* buf_p0 = smem + wave * (TILE_F4 * 4);
  const float* buf_p2 = buf_p0 + 2 * NWAVE * (TILE_F4 * 4);

  float m = -INFINITY, s = 0.0f;

  // ---- prime the pipeline: tiles 0 and 1 (8 async ops in flight) ----
  ISSUE_TILE(buf_b0, gbase);
  ISSUE_TILE(buf_b1, gbase + TILE_BYTES);

  // ---- tail (direct global loads; overlaps tile-0/1 arrival) ----
  {
    const float4* gx4 = (const float4*)rowp;
    for (int i = TAIL_START + lane; i < WAVE_F4; i += LANES) {
      float4 v = gx4[wave * WAVE_F4 + i];
      lse_update(m, s, v.x); lse_update(m, s, v.y);
      lse_update(m, s, v.z); lse_update(m, s, v.w);
    }
  }

  // ---- 3-deep async ring over 24 tiles ----
  uint32_t     ib = buf_b2;   // LDS base receiving tile t+2
  const float* pc = buf_p0;   // LDS region holding tile t
  for (int t = 0; t < NT; ++t) {
    if (t + 2 < NT) {
      // WAR guard: ds_loads from the buffer we are about to refill (iter t-1)
      asm volatile("s_wait_dscnt 0x0" ::: "memory");
      ISSUE_TILE(ib, gbase + (uint32_t)((t + 2) * TILE_BYTES));
      // tiles t+1, t+2 (8 ops) stay in flight; tile t complete (in-order done)
      asm volatile("s_wait_asynccnt 0x8" ::: "memory");
      ib = (ib == buf_b2) ? buf_b0 : ib + (uint32_t)(NWAVE * TILE_BYTES);
    } else if (t + 1 < NT) {
      asm volatile("s_wait_asynccnt 0x4" ::: "memory");
    } else {
      asm volatile("s_wait_asynccnt 0x0" ::: "memory");
    }

    const float4* sb = (const float4*)pc;
#pragma unroll
    for (int u = 0; u < 4; ++u) {
      float4 v = sb[u * 32 + lane];          // ds_load_b128
      lse_update(m, s, v.x); lse_update(m, s, v.y);
      lse_update(m, s, v.z); lse_update(m, s, v.w);
    }
    pc = (pc == buf_p2) ? buf_p0 : pc + NWAVE * (TILE_F4 * 4);
  }

  // ---- wave32 reduction of (m, s) ----
#pragma unroll
  for (int off = 16; off > 0; off >>= 1) {
    float mo = __shfl_xor(m, off, 32);
    float so = __shfl_xor(s, off, 32);
    lse_merge(m, s, mo, so);
  }
  if (lane == 0) { wm[wave] = m; wss[wave] = s; }
  __syncthreads();

  // ---- per-row fixup: excluded-set subtraction + rel loss (thread 0) ----
  if (tid == 0) {
    float M = wm[0], S = wss[0];
    for (int w = 1; w < NWAVE; ++w) lse_merge(M, S, wm[w], wss[w]);

    const int* yr = y + row * T_DIM;
    int   target  = yr[0];
    float x_tgt   = rowp[target];

    int idx[T_DIM]; int n = 0;
    idx[n++] = target;
    float xrel = INFINITY; int nvalid = 0;
    for (int t = 1; t < T_DIM; ++t) {
      int r = yr[t];
      if (r >= 0) {
        ++nvalid;
        float v = rowp[r];
        xrel = fminf(xrel, v);
        bool dup = false;
        for (int i2 = 0; i2 < n; ++i2) dup = dup || (idx[i2] == r);
        if (!dup) idx[n++] = r;
      }
    }
    float ex = 0.0f;
    for (int i2 = 0; i2 < n; ++i2) ex += fexp(rowp[idx[i2]] - M);

    float loss1 = (M + flog(S)) - x_tgt;
    float rel = 0.0f;
    if (nvalid >= 1) {
      float sc = fmaxf(S - ex, 1e-37f);
      float lse_cand = M + flog(sc);
      float mx = fmaxf(lse_cand, xrel);
      float laddexp = mx + log1pf(fexp(-fabsf(lse_cand - xrel)));
      rel = laddexp - xrel;
    }
    ws[row]         = loss1;
    ws[B + row]     = rel;
    ws[2 * B + row] = (nvalid >= 1) ? 1.0f : 0.0f;
  }
}

extern "C" __global__ void __launch_bounds__(256)
rll_finalize_kernel(const float* __restrict__ ws, float* __restrict__ out, int B) {
  __shared__ float a[256], b[256], c[256];
  int tid = threadIdx.x;
  float s1 = 0.0f, s2 = 0.0f, s3 = 0.0f;
  for (int i = tid; i < B; i += 256) {   // fixed order -> deterministic
    s1 += ws[i]; s2 += ws[B + i]; s3 += ws[2 * B + i];
  }
  a[tid] = s1; b[tid] = s2; c[tid] = s3;
  __syncthreads();
  for (int off = 128; off > 0; off >>= 1) {
    if (tid < off) { a[tid] += a[tid + off]; b[tid] += b[tid + off]; c[tid] += c[tid + off]; }
    __syncthreads();
  }
  if (tid == 0) out[0] = a[0] / (float)B + GAMMA_F * (b[0] / (1e-8f + c[0]));
}

extern "C" void kernel_launch(void* const* d_in, const int* in_sizes, int n_in,
                              void* d_out, int out_size, void* d_ws, size_t ws_size,
                              hipStream_t stream) {
  const float* x = (const float*)d_in[0];
  const int*   y = (const int*)d_in[1];
  float* ws  = (float*)d_ws;
  float* out = (float*)d_out;
  int B = in_sizes[0] / C_DIM;   // 1024

  rll_row_kernel<<<B, BLOCK, 0, stream>>>(x, y, ws, B);
  rll_finalize_kernel<<<1, 256, 0, stream>>>(ws, out, B);
}